// GraphSAGEPlusPlusDAC_89635967468051
// MI455X (gfx1250) — compile-verified
//
#include <hip/hip_runtime.h>
#include <hip/hip_bf16.h>

// ---------------------------------------------------------------------------
// GraphSAGE 2-layer + MLP head for MI455X (gfx1250, wave32).
//
// Problem sizes (compile-time constants from the reference):
static constexpr int N0 = 1171456, N1 = 45056, N2 = 4096;
static constexpr int E0 = 1126400, E1 = 40960;
static constexpr int DH = 128;     // D_IN == D_H == 128
static constexpr int DOUT = 47;
// ---------------------------------------------------------------------------

typedef __attribute__((ext_vector_type(2))) float v2f;
typedef __attribute__((ext_vector_type(8))) float v8f;

// ---------------------------------------------------------------------------
// Zero the accumulator + count regions (grid-stride float4 stores).
// ---------------------------------------------------------------------------
__global__ void zero_f4_kernel(float4* __restrict__ p, int n4) {
    int i = blockIdx.x * blockDim.x + threadIdx.x;
    int stride = gridDim.x * blockDim.x;
    for (; i < n4; i += stride) p[i] = make_float4(0.f, 0.f, 0.f, 0.f);
}

// ---------------------------------------------------------------------------
// Edge scatter: one wave per edge. Lane l moves feat[src][4l..4l+3] into
// acc[dst][4l..4l+3] via f32 atomics (accumulators are ~23MB / 2MB -> L2
// resident, so atomics resolve at L2). Lane 0 bumps the degree count.
// blockDim.x = 256 -> 8 edges per block.
// ---------------------------------------------------------------------------
__global__ void edge_scatter_kernel(const float* __restrict__ feat,
                                    const int* __restrict__ src,
                                    const int* __restrict__ dst,
                                    float* __restrict__ acc,
                                    float* __restrict__ cnt,
                                    int nEdges) {
    int lane = threadIdx.x & 31;
    int e = blockIdx.x * 8 + (threadIdx.x >> 5);
    if (e >= nEdges) return;
    int s = src[e];
    int d = dst[e];
    const float4 v = *(const float4*)(feat + (size_t)s * DH + lane * 4);
    float* a = acc + (size_t)d * DH + lane * 4;
    atomicAdd(a + 0, v.x);
    atomicAdd(a + 1, v.y);
    atomicAdd(a + 2, v.z);
    atomicAdd(a + 3, v.w);
    if (lane == 0) atomicAdd(cnt + d, 1.0f);
}

// ---------------------------------------------------------------------------
// Node update GEMM via fp32 WMMA (V_WMMA_F32_16X16X4_F32):
//   out = act( (acc/max(cnt,1)) @ Wl  +  self @ Wr  + bias )
// One wave computes one 16x16 output tile; K=128 stepped by 4 per wmma.
// M (rows) and the tile grid are exact multiples -> no divergence, EXEC
// stays all-ones around every WMMA as the ISA requires.
//
// Fragment layouts (ISA 7.12.2, 32-bit):
//   A 16x4 : lane(L) -> row = L&15, v2f = A[row][k + 2*(L>>4) .. +1]
//   B 4x16 : lane(L) -> col = L&15, v2f = {B[k+2*(L>>4)][col], B[k+2*(L>>4)+1][col]}
//   C 16x16: vgpr r  -> row = r + 8*(L>>4), col = L&15
// ---------------------------------------------------------------------------
template <bool RELU>
__global__ void node_gemm_kernel(const float* __restrict__ acc,
                                 const float* __restrict__ cnt,
                                 const float* __restrict__ self,
                                 const float* __restrict__ Wl,
                                 const float* __restrict__ Wr,
                                 const float* __restrict__ bias,
                                 float* __restrict__ out) {
    const int lane = threadIdx.x & 31;
    const int wave = threadIdx.x >> 5;
    const int tile = blockIdx.x * 8 + wave;   // blockDim.x == 256 -> 8 waves
    const int nT = tile & 7;                  // 128/16 = 8 tiles along N
    const int mT = tile >> 3;
    const int hi = lane >> 4;                 // half-wave select (K offset)
    const int lo = lane & 15;
    const int m0 = mT * 16;
    const int n0 = nT * 16;

    const int arow = m0 + lo;                 // A row owned by this lane
    const float inv = 1.0f / fmaxf(cnt[arow], 1.0f);

    v8f c = {};

    // GEMM 1: (acc * inv) @ Wl
    const float* arp = acc + (size_t)arow * DH;
    #pragma unroll
    for (int k = 0; k < DH; k += 4) {
        const int ka = k + 2 * hi;
        v2f a = *(const v2f*)(arp + ka);
        a.x *= inv; a.y *= inv;
        v2f b;
        b.x = Wl[(size_t)ka * DH + n0 + lo];
        b.y = Wl[(size_t)(ka + 1) * DH + n0 + lo];
        c = __builtin_amdgcn_wmma_f32_16x16x4_f32(false, a, false, b,
                                                  (short)0, c, false, false);
    }
    // GEMM 2: self @ Wr   (accumulate into same C)
    const float* srp = self + (size_t)arow * DH;
    #pragma unroll
    for (int k = 0; k < DH; k += 4) {
        const int ka = k + 2 * hi;
        v2f a = *(const v2f*)(srp + ka);
        v2f b;
        b.x = Wr[(size_t)ka * DH + n0 + lo];
        b.y = Wr[(size_t)(ka + 1) * DH + n0 + lo];
        c = __builtin_amdgcn_wmma_f32_16x16x4_f32(false, a, false, b,
                                                  (short)0, c, false, false);
    }

    const float bv = bias[n0 + lo];
    #pragma unroll
    for (int r = 0; r < 8; ++r) {
        const int row = m0 + r + 8 * hi;
        float v = c[r] + bv;
        if (RELU) v = fmaxf(v, 0.0f);
        out[(size_t)row * DH + n0 + lo] = v;
    }
}

// ---------------------------------------------------------------------------
// Head: out = log_softmax(h2 @ Wp + bp). One block (64 threads) per row.
// ---------------------------------------------------------------------------
__global__ void post_kernel(const float* __restrict__ h2,
                            const float* __restrict__ Wp,
                            const float* __restrict__ bp,
                            float* __restrict__ out) {
    __shared__ float logits[64];
    __shared__ float red[64];
    const int row = blockIdx.x;
    const int j = threadIdx.x;                // blockDim.x == 64

    float lg = -__builtin_inff();
    if (j < DOUT) {
        float s = bp[j];
        const float* hr = h2 + (size_t)row * DH;
        #pragma unroll 8
        for (int k = 0; k < DH; ++k) s += hr[k] * Wp[k * DOUT + j];
        lg = s;
    }
    logits[j] = lg;
    red[j] = lg;
    __syncthreads();
    for (int off = 32; off > 0; off >>= 1) {
        if (j < off) red[j] = fmaxf(red[j], red[j + off]);
        __syncthreads();
    }
    const float mx = red[0];
    __syncthreads();
    red[j] = (j < DOUT) ? __expf(logits[j] - mx) : 0.0f;
    __syncthreads();
    for (int off = 32; off > 0; off >>= 1) {
        if (j < off) red[j] += red[j + off];
        __syncthreads();
    }
    const float lse = mx + __logf(red[0]);
    if (j < DOUT) out[(size_t)row * DOUT + j] = logits[j] - lse;
}

// ---------------------------------------------------------------------------
extern "C" void kernel_launch(void* const* d_in, const int* in_sizes, int n_in,
                              void* d_out, int out_size, void* d_ws, size_t ws_size,
                              hipStream_t stream) {
    (void)in_sizes; (void)n_in; (void)out_size; (void)ws_size;

    const float* x    = (const float*)d_in[0];
    const int*   src0 = (const int*)  d_in[1];
    const int*   dst0 = (const int*)  d_in[2];
    const int*   src1 = (const int*)  d_in[3];
    const int*   dst1 = (const int*)  d_in[4];
    const float* Wl0  = (const float*)d_in[5];
    const float* bl0  = (const float*)d_in[6];
    const float* Wr0  = (const float*)d_in[7];
    const float* Wl1  = (const float*)d_in[8];
    const float* bl1  = (const float*)d_in[9];
    const float* Wr1  = (const float*)d_in[10];
    const float* Wp   = (const float*)d_in[11];
    const float* bp   = (const float*)d_in[12];
    float* out = (float*)d_out;

    // Workspace layout (floats): acc0 | cnt0 | acc1 | cnt1 | h | h2
    float* acc0 = (float*)d_ws;                   // N1*128
    float* cnt0 = acc0 + (size_t)N1 * DH;         // N1
    float* acc1 = cnt0 + N1;                      // N2*128
    float* cnt1 = acc1 + (size_t)N2 * DH;         // N2
    float* h    = cnt1 + N2;                      // N1*128
    float* h2   = h + (size_t)N1 * DH;            // N2*128

    // 1) zero accumulators + counts (must happen every call: atomics below)
    const int nZero4 = (N1 * DH + N1 + N2 * DH + N2) / 4;   // 1,585,152
    zero_f4_kernel<<<6192, 256, 0, stream>>>((float4*)d_ws, nZero4);

    // 2) layer-0 edge scatter (bandwidth-bound path: ~0.6 GB of HBM traffic)
    edge_scatter_kernel<<<E0 / 8, 256, 0, stream>>>(x, src0, dst0, acc0, cnt0, E0);

    // 3) layer-0 node GEMM + ReLU: 45056/16 = 2816 M-tiles x 8 N-tiles, /8 waves
    node_gemm_kernel<true><<<(N1 / 16) * 8 / 8, 256, 0, stream>>>(
        acc0, cnt0, x /* x[:N1] shares the base */, Wl0, Wr0, bl0, h);

    // 4) layer-1 edge scatter
    edge_scatter_kernel<<<E1 / 8, 256, 0, stream>>>(h, src1, dst1, acc1, cnt1, E1);

    // 5) layer-1 node GEMM (no ReLU)
    node_gemm_kernel<false><<<(N2 / 16) * 8 / 8, 256, 0, stream>>>(
        acc1, cnt1, h /* h[:N2] shares the base */, Wl1, Wr1, bl1, h2);

    // 6) MLP head + log_softmax
    post_kernel<<<N2, 64, 0, stream>>>(h2, Wp, bp, out);
}